// CapsuleLayer_16638703305145
// MI455X (gfx1250) — compile-verified
//
#include <hip/hip_runtime.h>
#include <hip/hip_bf16.h>
#include <math.h>

typedef __attribute__((ext_vector_type(2))) float v2f;
typedef __attribute__((ext_vector_type(4))) float v4f;
typedef __attribute__((ext_vector_type(8))) float v8f;

#define BATCH      128
#define IN_UNIT    16
#define IN_CH      1152
#define NUM_UNIT   16
#define UNIT_SIZE  16
#define NUM_ROUT   3

// -----------------------------------------------------------------------------
// Kernel 1: u_hat[b,c,n,s] = sum_i W[c,n,s,i] * x[b,i,c]
// One block = one (channel c, 16-batch tile). 512 threads = 16 wave32, wave w
// computes capsule n=w via 4 chained V_WMMA_F32_16X16X4_F32 (M=s, N=b, K=i).
// -----------------------------------------------------------------------------
__global__ __launch_bounds__(512)
void uhat_wmma_kernel(const float* __restrict__ x,    // [B, IN_UNIT, IN_CH]
                      const float* __restrict__ w,    // [C, N, S, I]
                      float* __restrict__ uhat)       // [B, C, N, S]
{
    const int c     = blockIdx.x;
    const int btile = blockIdx.y;           // 0..7 : batches btile*16 .. +15
    const int t     = threadIdx.x;

    __shared__ float xt[IN_UNIT][16];       // xt[i][b_local] = x[b, i, c]
    if (t < 256) {
        const int i  = t >> 4;
        const int bl = t & 15;
        xt[i][bl] = x[(size_t)(btile * 16 + bl) * (IN_UNIT * IN_CH)
                      + (size_t)i * IN_CH + c];
    }
    __syncthreads();

    const int n    = t >> 5;                // wave id == capsule index n
    const int lane = t & 31;
    const int half = lane >> 4;             // 0: lanes 0-15, 1: lanes 16-31
    const int l15  = lane & 15;

    // A-fragment source: W[c][n][s=l15][i], K dimension = i
    const float* wa = w + (((size_t)c * NUM_UNIT + n) * UNIT_SIZE + l15) * IN_UNIT;

    v8f acc = {};
    #pragma unroll
    for (int k0 = 0; k0 < IN_UNIT; k0 += 4) {
        const int ka = k0 + 2 * half;       // A 16x4 f32 layout: lane-half holds K pair
        v2f a;
        a[0] = wa[ka];
        a[1] = wa[ka + 1];
        v2f bf;                              // B 4x16 f32 layout (mirrors A K-split)
        bf[0] = xt[ka][l15];
        bf[1] = xt[ka + 1][l15];
        acc = __builtin_amdgcn_wmma_f32_16x16x4_f32(
            /*neg_a=*/false, a, /*neg_b=*/false, bf,
            /*c_mod=*/(short)0, acc, /*reuse_a=*/false, /*reuse_b=*/false);
    }

    // D layout: VGPR r -> s = r (lanes 0-15) or r+8 (lanes 16-31); N = b_local.
    // Each lane owns 8 consecutive s values -> two coalesced b128 stores.
    const int b  = btile * 16 + l15;
    const int s0 = half * 8;
    float* dst = uhat + ((size_t)b * IN_CH + c) * (NUM_UNIT * UNIT_SIZE)
                      + n * UNIT_SIZE + s0;
    v4f lo, hi;
    lo[0] = acc[0]; lo[1] = acc[1]; lo[2] = acc[2]; lo[3] = acc[3];
    hi[0] = acc[4]; hi[1] = acc[5]; hi[2] = acc[6]; hi[3] = acc[7];
    *(v4f*)(dst)     = lo;
    *(v4f*)(dst + 4) = hi;
}

// -----------------------------------------------------------------------------
// Zero routing logits b_ij each launch (deterministic across replays).
// -----------------------------------------------------------------------------
__global__ void zero_bij_kernel(float* __restrict__ bij)
{
    const int i = blockIdx.x * blockDim.x + threadIdx.x;
    if (i < IN_CH * NUM_UNIT) bij[i] = 0.0f;
}

// -----------------------------------------------------------------------------
// c_ij = softmax(b_ij) over num_unit, one thread per channel row (16 values).
// -----------------------------------------------------------------------------
__global__ void softmax_kernel(const float* __restrict__ bij,
                               float* __restrict__ cij)
{
    const int c = blockIdx.x * blockDim.x + threadIdx.x;
    if (c >= IN_CH) return;
    float v[NUM_UNIT];
    float m = -INFINITY;
    #pragma unroll
    for (int n = 0; n < NUM_UNIT; ++n) {
        v[n] = bij[c * NUM_UNIT + n];
        m = fmaxf(m, v[n]);
    }
    float s = 0.0f;
    #pragma unroll
    for (int n = 0; n < NUM_UNIT; ++n) {
        v[n] = __expf(v[n] - m);
        s += v[n];
    }
    const float inv = 1.0f / s;
    #pragma unroll
    for (int n = 0; n < NUM_UNIT; ++n)
        cij[c * NUM_UNIT + n] = v[n] * inv;
}

// -----------------------------------------------------------------------------
// s_j[b,n,:] = sum_c c_ij[c,n] * u_hat[b,c,n,:]; v_j = squash(s_j).
// One block per (b,n); 256 threads stride over C with b128 u_hat reads.
// -----------------------------------------------------------------------------
__global__ __launch_bounds__(256)
void sj_squash_kernel(const float* __restrict__ uhat,
                      const float* __restrict__ cij,
                      float* __restrict__ vj,
                      float* __restrict__ out,
                      int write_out)
{
    const int b = blockIdx.x;
    const int n = blockIdx.y;
    const int t = threadIdx.x;

    float acc[UNIT_SIZE];
    #pragma unroll
    for (int s = 0; s < UNIT_SIZE; ++s) acc[s] = 0.0f;

    for (int c = t; c < IN_CH; c += 256) {
        const float wgt = cij[c * NUM_UNIT + n];
        const float* up = uhat + ((size_t)b * IN_CH + c) * (NUM_UNIT * UNIT_SIZE)
                               + n * UNIT_SIZE;
        #pragma unroll
        for (int s4 = 0; s4 < UNIT_SIZE; s4 += 4) {
            const v4f u = *(const v4f*)(up + s4);
            acc[s4 + 0] += wgt * u[0];
            acc[s4 + 1] += wgt * u[1];
            acc[s4 + 2] += wgt * u[2];
            acc[s4 + 3] += wgt * u[3];
        }
    }

    __shared__ float red[256][UNIT_SIZE + 1];   // +1 pad: avoid bank conflicts
    #pragma unroll
    for (int s = 0; s < UNIT_SIZE; ++s) red[t][s] = acc[s];
    __syncthreads();
    for (int stride = 128; stride > 0; stride >>= 1) {
        if (t < stride) {
            #pragma unroll
            for (int s = 0; s < UNIT_SIZE; ++s)
                red[t][s] += red[t + stride][s];
        }
        __syncthreads();
    }

    __shared__ float coef_sh;
    if (t == 0) {
        float sq = 0.0f;
        #pragma unroll
        for (int s = 0; s < UNIT_SIZE; ++s) {
            const float v = red[0][s];
            sq += v * v;
        }
        const float norm = sqrtf(sq);
        coef_sh = sq / ((1.0f + sq) * norm);   // squash: |s|^2/(1+|s|^2) / |s|
    }
    __syncthreads();

    if (t < UNIT_SIZE) {
        const float val = coef_sh * red[0][t];
        const size_t idx = ((size_t)b * NUM_UNIT + n) * UNIT_SIZE + t;
        vj[idx] = val;
        if (write_out) out[idx] = val;         // output layout [B,N,S,1] == flat
    }
}

// -----------------------------------------------------------------------------
// b_ij[c,n] += mean_b sum_s u_hat[b,c,n,s] * v_j[b,n,s].
// One block per (c,n); 128 threads = one per batch element.
// -----------------------------------------------------------------------------
__global__ __launch_bounds__(128)
void agree_kernel(const float* __restrict__ uhat,
                  const float* __restrict__ vj,
                  float* __restrict__ bij)
{
    const int c = blockIdx.x;
    const int n = blockIdx.y;
    const int b = threadIdx.x;   // 0..127

    const float* up = uhat + ((size_t)b * IN_CH + c) * (NUM_UNIT * UNIT_SIZE)
                           + n * UNIT_SIZE;
    const float* vp = vj + ((size_t)b * NUM_UNIT + n) * UNIT_SIZE;

    float dot = 0.0f;
    #pragma unroll
    for (int s4 = 0; s4 < UNIT_SIZE; s4 += 4) {
        const v4f u = *(const v4f*)(up + s4);
        const v4f v = *(const v4f*)(vp + s4);
        dot += u[0] * v[0] + u[1] * v[1] + u[2] * v[2] + u[3] * v[3];
    }

    __shared__ float red[128];
    red[b] = dot;
    __syncthreads();
    for (int st = 64; st > 0; st >>= 1) {
        if (b < st) red[b] += red[b + st];
        __syncthreads();
    }
    if (b == 0) bij[c * NUM_UNIT + n] += red[0] * (1.0f / (float)BATCH);
}

// -----------------------------------------------------------------------------
extern "C" void kernel_launch(void* const* d_in, const int* in_sizes, int n_in,
                              void* d_out, int out_size, void* d_ws, size_t ws_size,
                              hipStream_t stream)
{
    const float* x = (const float*)d_in[0];     // [B, IN_UNIT, IN_CH]
    // d_in[1] = x_original (unused), d_in[3] = mode, d_in[4] = epoch (unused)
    const float* w = (const float*)d_in[2];     // [C, N, S, I]
    float* out = (float*)d_out;                 // [B, N, S, 1] flat

    const size_t uhat_elems = (size_t)BATCH * IN_CH * NUM_UNIT * UNIT_SIZE;
    float* uhat = (float*)d_ws;                         // 151 MB, L2-resident
    float* bij  = uhat + uhat_elems;                    // [C, N]
    float* cij  = bij + (size_t)IN_CH * NUM_UNIT;       // [C, N]
    float* vj   = cij + (size_t)IN_CH * NUM_UNIT;       // [B, N, S]

    dim3 g1(IN_CH, BATCH / 16);
    uhat_wmma_kernel<<<g1, 512, 0, stream>>>(x, w, uhat);

    zero_bij_kernel<<<(IN_CH * NUM_UNIT + 255) / 256, 256, 0, stream>>>(bij);

    for (int it = 0; it < NUM_ROUT; ++it) {
        softmax_kernel<<<(IN_CH + 255) / 256, 256, 0, stream>>>(bij, cij);

        dim3 g2(BATCH, NUM_UNIT);
        sj_squash_kernel<<<g2, 256, 0, stream>>>(uhat, cij, vj, out,
                                                 it == NUM_ROUT - 1 ? 1 : 0);

        if (it < NUM_ROUT - 1) {
            dim3 g3(IN_CH, NUM_UNIT);
            agree_kernel<<<g3, 128, 0, stream>>>(uhat, vj, bij);
        }
    }
}